// MedianFilter_21998822490112
// MI455X (gfx1250) — compile-verified
//
#include <hip/hip_runtime.h>
#include <math.h>

// Problem constants (reference: image (3, 1024, 1024) fp32, 5x5 masked median)
#define C_DIM 3
#define H_DIM 1024
#define W_DIM 1024

#define TILE_W 32
#define TILE_H 16
#define HALO_W (TILE_W + 4)             // 36
#define HALO_H (TILE_H + 4)             // 20
#define LDS_ELEMS (HALO_W * HALO_H)     // 720
#define NTHREADS 256
#define STAGE_ITERS ((LDS_ELEMS + NTHREADS - 1) / NTHREADS)

// ---------------------------------------------------------------------------
// Compile-time sorting network: Batcher merge-exchange (Knuth 5.2.2M) for
// n=25 (138 CEs), backward-pruned to the cone of influence of a caller-
// specified set of sorted output positions:
//   - interior tiles need only position 12 (true median of 25)
//   - border tiles need positions 0..12 (loIdx=(n-1)/2, hiIdx=n/2, n in [4,25])
// ---------------------------------------------------------------------------
struct Net {
    unsigned char a[160];
    unsigned char b[160];
    int cnt;
};

constexpr Net build_net(unsigned needed_mask) {
    constexpr int n = 25, t = 5;      // 2^5 = 32 >= 25
    Net full{};
    int c = 0;
    for (int p = 1 << (t - 1); p > 0; p >>= 1) {
        int q = 1 << (t - 1), r = 0, d = p;
        for (;;) {
            for (int i = 0; i < n - d; ++i)
                if ((i & p) == r) {
                    full.a[c] = (unsigned char)i;
                    full.b[c] = (unsigned char)(i + d);
                    ++c;
                }
            if (q == p) break;
            d = q - p; q >>= 1; r = p;
        }
    }
    full.cnt = c;

    // Backward slice: keep a CE iff it can influence a needed output position.
    bool keep[160] = {};
    unsigned needed = needed_mask;
    for (int i = c - 1; i >= 0; --i) {
        const unsigned m = (1u << full.a[i]) | (1u << full.b[i]);
        if (needed & m) { keep[i] = true; needed |= m; }
    }
    Net out{};
    int oc = 0;
    for (int i = 0; i < c; ++i)
        if (keep[i]) { out.a[oc] = full.a[i]; out.b[oc] = full.b[i]; ++oc; }
    out.cnt = oc;
    return out;
}

template <unsigned NEEDED_MASK>
__device__ __forceinline__ void sort_net(float v[25]) {
    constexpr Net net = build_net(NEEDED_MASK);
    #pragma unroll
    for (int s = 0; s < net.cnt; ++s) {
        const int ia = net.a[s];
        const int ib = net.b[s];
        const float x = v[ia];
        const float y = v[ib];
        v[ia] = fminf(x, y);
        v[ib] = fmaxf(x, y);
    }
}

__device__ __forceinline__ void gather25(const float* tile, int ly, int lx,
                                         float v[25]) {
    #pragma unroll
    for (int dy = 0; dy < 5; ++dy) {
        #pragma unroll
        for (int dx = 0; dx < 5; ++dx) {
            v[dy * 5 + dx] = tile[(ly + dy) * HALO_W + (lx + dx)];
        }
    }
}

__global__ __launch_bounds__(NTHREADS)
void median5x5_masked_kernel(const float* __restrict__ img,
                             float* __restrict__ out) {
    __shared__ float tile[LDS_ELEMS];

    const int tid = threadIdx.x;
    const int tx0 = blockIdx.x * TILE_W;
    const int ty0 = blockIdx.y * TILE_H;
    const int c   = blockIdx.z;
    const float* __restrict__ src = img + (size_t)c * H_DIM * W_DIM;

    // ---------------------------------------------------------------
    // Stage halo tile into LDS via CDNA5 async global->LDS copies
    // (ASYNCcnt-tracked). Coordinates are clamped so OOB taps read a
    // valid address; invalid entries are overwritten with +inf below.
    // ---------------------------------------------------------------
    #pragma unroll
    for (int it = 0; it < STAGE_ITERS; ++it) {
        int idx = tid + it * NTHREADS;
        idx = idx < LDS_ELEMS ? idx : (LDS_ELEMS - 1);   // keep EXEC full; dup write benign
        const int ly = idx / HALO_W;
        const int lx = idx - ly * HALO_W;
        int gy = ty0 + ly - 2; gy = gy < 0 ? 0 : (gy > H_DIM - 1 ? H_DIM - 1 : gy);
        int gx = tx0 + lx - 2; gx = gx < 0 ? 0 : (gx > W_DIM - 1 ? W_DIM - 1 : gx);
        const unsigned long long gaddr =
            (unsigned long long)(src + (size_t)gy * W_DIM + gx);
        const unsigned lds_off = (unsigned)(size_t)(&tile[idx]);  // low 32b = group-segment offset
        asm volatile("global_load_async_to_lds_b32 %0, %1, off"
                     :: "v"(lds_off), "v"(gaddr)
                     : "memory");
    }
    asm volatile("s_wait_asynccnt 0" ::: "memory");

    // Block-uniform classification: interior tiles have every window tap
    // valid (n==25 for all pixels); only border tiles carry masked taps.
    const bool border_tile = (tx0 == 0) || (ty0 == 0) ||
                             (tx0 + TILE_W + 1 > W_DIM - 2) ||
                             (ty0 + TILE_H + 1 > H_DIM - 2);

    // Poison invalid taps with +inf (mask depends only on absolute coords,
    // not the window center). Only border tiles have any; uniform branch.
    if (border_tile) {
        #pragma unroll
        for (int it = 0; it < STAGE_ITERS; ++it) {
            const int idx = tid + it * NTHREADS;
            if (idx < LDS_ELEMS) {
                const int ly = idx / HALO_W;
                const int lx = idx - ly * HALO_W;
                const int y = ty0 + ly - 2;
                const int x = tx0 + lx - 2;
                const bool valid = ((unsigned)y <= (unsigned)(H_DIM - 2)) &&
                                   ((unsigned)x <= (unsigned)(W_DIM - 2));
                if (!valid) tile[idx] = __builtin_inff();
            }
        }
    }
    __syncthreads();

    // ---------------------------------------------------------------
    // Each thread computes 2 output pixels (32x8 thread layout, 2 rows)
    // ---------------------------------------------------------------
    const int lx = tid & (TILE_W - 1);   // 0..31 (one wave spans a tile row)
    const int lyBase = tid >> 5;         // 0..7
    float* const dst = out + (size_t)c * H_DIM * W_DIM;

    if (!border_tile) {
        // Common path (~91% of tiles): n==25 everywhere -> output is the
        // single order statistic at sorted position 12. Median-only pruned
        // network; no n computation, no selection chain.
        #pragma unroll
        for (int r = 0; r < TILE_H / 8; ++r) {
            const int ly = lyBase + r * 8;
            const int gy = ty0 + ly;
            const int gx = tx0 + lx;

            float v[25];
            gather25(tile, ly, lx, v);
            sort_net<(1u << 12)>(v);
            dst[(size_t)gy * W_DIM + gx] = v[12];
        }
    } else {
        // Border path: masked taps are +inf and sink out of the selected
        // range; pick order statistics (n-1)/2 and n/2 (both <= 12).
        #pragma unroll
        for (int r = 0; r < TILE_H / 8; ++r) {
            const int ly = lyBase + r * 8;
            const int gy = ty0 + ly;
            const int gx = tx0 + lx;

            float v[25];
            gather25(tile, ly, lx, v);

            const int yl = (gy - 2) > 0 ? (gy - 2) : 0;
            const int yh = (gy + 2) < (H_DIM - 2) ? (gy + 2) : (H_DIM - 2);
            const int xl = (gx - 2) > 0 ? (gx - 2) : 0;
            const int xh = (gx + 2) < (W_DIM - 2) ? (gx + 2) : (W_DIM - 2);
            const int n  = (yh - yl + 1) * (xh - xl + 1);    // 4..25

            sort_net<(1u << 13) - 1>(v);   // positions 0..12 correct

            const int loIdx = (n - 1) >> 1;
            const int hiIdx = n >> 1;
            float lo = v[0];
            float hi = v[0];
            #pragma unroll
            for (int i = 1; i <= 12; ++i) {
                lo = (i == loIdx) ? v[i] : lo;
                hi = (i == hiIdx) ? v[i] : hi;
            }
            dst[(size_t)gy * W_DIM + gx] = 0.5f * (lo + hi);
        }
    }
}

extern "C" void kernel_launch(void* const* d_in, const int* in_sizes, int n_in,
                              void* d_out, int out_size, void* d_ws, size_t ws_size,
                              hipStream_t stream) {
    (void)in_sizes; (void)n_in; (void)out_size; (void)d_ws; (void)ws_size;
    const float* img = (const float*)d_in[0];
    float* out = (float*)d_out;

    dim3 block(NTHREADS, 1, 1);
    dim3 grid(W_DIM / TILE_W, H_DIM / TILE_H, C_DIM);
    median5x5_masked_kernel<<<grid, block, 0, stream>>>(img, out);
}